// ExactScalarArray_86483461472728
// MI455X (gfx1250) — compile-verified
//
#include <hip/hip_runtime.h>

// ---------------------------------------------------------------------------
// ExactScalarArray for MI455X (gfx1250, wave32).  B=256, L=2048, K=8.
//
// Roofline: 160 MiB read once / 2 KiB out -> ~7 us at 23.3 TB/s; VALU body
// (~240 FMA-class ops per (b,l)) is ~4 us — so both data movement and VALU
// count matter.  Decisions:
//  * 1024 blocks (4 L-chunks per batch) x 8 waves = 8192 waves so that
//    outstanding b128 loads cover the ~14 MB BW*latency product of HBM.
//  * Non-temporal (TH=NT) loads: data is streamed exactly once; don't churn
//    the 192 MB L2.
//  * while-loop -> sh = ctz(OR of int patterns); halve via exact *2^-sh in
//    f32 (no int->float converts).  minp cancels algebraically -> one pass.
//  * Per-block reduction finishes with V_WMMA_F32_16X16X4_F32:
//    A = ones(16x4), B = 8x4 wave-partials packed so each column's 4
//    K-entries are one component -> D row 0 = exact f32 column sums.
//  * Tiny second kernel folds the 4 chunk-partials per batch and applies the
//    real/imag linear map.
// ---------------------------------------------------------------------------

#define INV_SQRT2 0.70710678118654752440f

typedef __attribute__((ext_vector_type(2))) float v2f;
typedef __attribute__((ext_vector_type(4))) float v4f;
typedef __attribute__((ext_vector_type(8))) float v8f;
typedef __attribute__((ext_vector_type(4))) int   v4i;

struct Q4 { float a, b, c, e; };

__device__ __forceinline__ Q4 smul(const Q4 d1, const Q4 d2) {
    Q4 r;
    r.a = d1.a * d2.a + d1.b * d2.e - d1.c * d2.c + d1.e * d2.b;
    r.b = d1.a * d2.b + d1.b * d2.a + d1.c * d2.e + d1.e * d2.c;
    r.c = d1.a * d2.c + d1.b * d2.b + d1.c * d2.a - d1.e * d2.e;
    r.e = d1.a * d2.e - d1.b * d2.c - d1.c * d2.b + d1.e * d2.a;
    return r;
}

__global__ __launch_bounds__(256) void esa_partial_kernel(
    const float* __restrict__ coeffs1, const float* __restrict__ coeffs2,
    const int* __restrict__ power1, const int* __restrict__ power2,
    float* __restrict__ ws) {

    const int bc    = blockIdx.x;     // 1024 = 256 batches x 4 L-chunks
    const int b     = bc >> 2;
    const int chunk = bc & 3;
    const int tid   = threadIdx.x;
    const int lane  = tid & 31;
    const int wv    = tid >> 5;

    __shared__ float s_part[8][4];    // 8 wave partials x 4 components

    float acc0 = 0.f, acc1 = 0.f, acc2 = 0.f, acc3 = 0.f;

    const int l0 = chunk * 512 + tid;
    #pragma unroll
    for (int it = 0; it < 2; ++it) {
        const int l = l0 + it * 256;
        const long base = (long)b * 2048 + l;
        const v4f* c1v = (const v4f*)(coeffs1 + base * 32);  // 8 x b128
        const v4f* c2v = (const v4f*)(coeffs2 + base * 32);
        const v4i* p1v = (const v4i*)(power1 + base * 8);    // 2 x b128
        const v4i* p2v = (const v4i*)(power2 + base * 8);

        if (it == 0) {   // global_prefetch for the second L-iteration
            __builtin_prefetch(coeffs1 + (base + 256) * 32, 0, 1);
            __builtin_prefetch(coeffs2 + (base + 256) * 32, 0, 1);
        }

        const v4i pa1 = __builtin_nontemporal_load(p1v);
        const v4i pb1 = __builtin_nontemporal_load(p1v + 1);
        const v4i pa2 = __builtin_nontemporal_load(p2v);
        const v4i pb2 = __builtin_nontemporal_load(p2v + 1);
        const int pw[8] = {pa1.x + pa2.x, pa1.y + pa2.y, pa1.z + pa2.z, pa1.w + pa2.w,
                           pb1.x + pb2.x, pb1.y + pb2.y, pb1.z + pb2.z, pb1.w + pb2.w};

        Q4 prod;
        int ppsum = 0;
        #pragma unroll
        for (int k = 0; k < 8; ++k) {
            const v4f a4 = __builtin_nontemporal_load(c1v + k);
            const v4f b4 = __builtin_nontemporal_load(c2v + k);
            const Q4 d1 = {a4.x, a4.y, a4.z, a4.w};
            const Q4 d2 = {b4.x, b4.y, b4.z, b4.w};
            const Q4 z = smul(d1, d2);

            // components are exact small integers: the reference's while-loop
            // equals one shift by sh = ctz of the OR'd bit patterns
            const int ia = (int)z.a, ib = (int)z.b, ic = (int)z.c, ie = (int)z.e;
            const unsigned m = (unsigned)(ia | ib | ic | ie);
            const int sh = m ? __builtin_ctz(m) : 0;
            // exact halving in f32: multiply by 2^-sh (even ints, no rounding)
            const float inv = __uint_as_float((unsigned)(127 - sh) << 23);
            const Q4 zr = {z.a * inv, z.b * inv, z.c * inv, z.e * inv};
            const int pk = pw[k] + sh;

            if (k == 0) { prod = zr; ppsum = pk; }
            else        { prod = smul(prod, zr); ppsum += pk; }
        }

        // 2^ppsum via exponent bits (0 <= ppsum <= ~32)
        const float scale = __uint_as_float((unsigned)(ppsum + 127) << 23);
        acc0 += prod.a * scale;
        acc1 += prod.b * scale;
        acc2 += prod.c * scale;
        acc3 += prod.e * scale;
    }

    // wave32 butterfly reduction
    #pragma unroll
    for (int off = 16; off; off >>= 1) {
        acc0 += __shfl_xor(acc0, off, 32);
        acc1 += __shfl_xor(acc1, off, 32);
        acc2 += __shfl_xor(acc2, off, 32);
        acc3 += __shfl_xor(acc3, off, 32);
    }
    if (lane == 0) {
        s_part[wv][0] = acc0; s_part[wv][1] = acc1;
        s_part[wv][2] = acc2; s_part[wv][3] = acc3;
    }
    __syncthreads();

    if (tid < 32) {   // wave 0, full EXEC (WMMA requirement)
        // B (4x16): column n's 4 K-entries come from lanes {n, n+16} x 2 VGPRs.
        // cols 0..3 = comps of partials 0..3; cols 4..7 = comps of partials 4..7.
        const int n  = lane & 15;
        const int hh = lane >> 4;
        float b0 = 0.f, b1 = 0.f;
        if (n < 8) {
            const int c = n & 3;
            const int g = n >> 2;
            const int w = 4 * g + 2 * hh;
            b0 = s_part[w][c];
            b1 = s_part[w + 1][c];
        }
        const v2f A  = {1.0f, 1.0f};   // ones(16x4)
        const v2f Bm = {b0, b1};
        v8f C = {};
        // D[m][n] = sum_k B[k][n]  (exact f32 column sums, all rows equal)
        v8f D = __builtin_amdgcn_wmma_f32_16x16x4_f32(
            false, A, false, Bm, (short)0, C, false, false);

        const float col = D[0];                          // row M=0, N=lane
        const float other = __shfl(col, (lane & 3) + 4, 32);
        const float S = col + other;                     // lanes 0..3: comp sums
        if (lane < 4) ws[bc * 4 + lane] = S;             // block partial -> ws
    }
}

// Fold the 4 chunk-partials per batch, apply the real/imag map (2^minp cancels).
__global__ __launch_bounds__(256) void esa_final_kernel(
    const float* __restrict__ ws, float* __restrict__ out) {
    const int b = threadIdx.x;                 // one block of 256 threads
    const v4f* p = (const v4f*)(ws + b * 16);  // 4 chunk partials x 4 comps
    const v4f s = p[0] + p[1] + p[2] + p[3];
    out[b * 2 + 0] = s.x + (s.y + s.w) * INV_SQRT2;
    out[b * 2 + 1] = s.z + (s.y - s.w) * INV_SQRT2;
}

extern "C" void kernel_launch(void* const* d_in, const int* in_sizes, int n_in,
                              void* d_out, int out_size, void* d_ws, size_t ws_size,
                              hipStream_t stream) {
    const float* c1 = (const float*)d_in[0];
    const float* c2 = (const float*)d_in[1];
    const int*   p1 = (const int*)d_in[2];
    const int*   p2 = (const int*)d_in[3];
    float* ws  = (float*)d_ws;   // 1024 blocks x 4 floats = 16 KB scratch
    float* out = (float*)d_out;

    esa_partial_kernel<<<dim3(1024), dim3(256), 0, stream>>>(c1, c2, p1, p2, ws);
    esa_final_kernel<<<dim3(1), dim3(256), 0, stream>>>(ws, out);
}